// QSHEncoding_70918499992406
// MI455X (gfx1250) — compile-verified
//
#include <hip/hip_runtime.h>
#include <hip/hip_bf16.h>
#include <stdint.h>

#define BLOCK 256

// ---------------------------------------------------------------------------
// Quantized SH-16, computed in a x1024 "scaled" fixed-point domain.
//   fl((2^k a)*b) == 2^k fl(a*b)  -> fold the x1024 of fpq into constants.
//   All clip(+-64) clamps are provably dead (|values| <= ~2.9 << 64).
//   Inner composites are exact integers in the scaled domain -> rint is id.
// Bit-exact with the reference under RTE (v_rndne_f32).
// ---------------------------------------------------------------------------
__device__ __forceinline__ void sh16(float i0, float i1, float i2, float* o) {
    const float K = 0.0009765625f;                      // 2^-10
    // scaled quantized features: xs = 1024 * fpq(2*i - 1)
    float xs = __builtin_rintf(fmaf(i0, 2048.0f, -1024.0f));
    float ys = __builtin_rintf(fmaf(i1, 2048.0f, -1024.0f));
    float zs = __builtin_rintf(fmaf(i2, 2048.0f, -1024.0f));
    float x = xs * K, y = ys * K, z = zs * K;           // exact unscale
    // scaled quantized products: e.g. xys = 1024 * fpq(x*y)
    float xys = __builtin_rintf(xs * y);
    float xzs = __builtin_rintf(xs * z);
    float yzs = __builtin_rintf(ys * z);
    float x2s = __builtin_rintf(xs * x);
    float y2s = __builtin_rintf(ys * y);
    float z2s = __builtin_rintf(zs * z);

    o[0] = 0.2822265625f;                               // qc(0.28209...) = 289/1024
    o[1] = __builtin_rintf(-500.0f * y) * K;            // qc(-0.48860)*1024 = -500
    o[2] = __builtin_rintf( 500.0f * z) * K;
    o[3] = __builtin_rintf(-500.0f * x) * K;
    o[4] = __builtin_rintf( 1.0927734375f * xys) * K;   // (1119/1024)*xys = 1119*xy
    o[5] = __builtin_rintf(-1.0927734375f * yzs) * K;
    o[6] = __builtin_rintf(fmaf(0.9462890625f, z2s, -323.0f)) * K; // 969*z2-323 (mul exact)
    o[7] = __builtin_rintf(-1.0927734375f * xzs) * K;
    o[8] = __builtin_rintf(fmaf(-0.5458984375f, y2s, 0.5458984375f * x2s)) * K;

    // inner qf() terms: exact integers in scaled domain (single fma each)
    float in9  = fmaf(x2s, -3.0f, y2s);                 // 1024*fpq(-3x2+y2)
    float in11 = fmaf(z2s, -5.0f, 1024.0f);             // 1024*fpq(1-5z2)
    float in12 = fmaf(z2s,  5.0f, -3072.0f);            // 1024*fpq(5z2-3)
    float in14 = x2s - y2s;                             // 1024*fpq(x2-y2)
    float in15 = fmaf(y2s, 3.0f, -x2s);                 // 1024*fpq(-x2+3y2)

    o[9]  = __builtin_rintf((0.58984375f  * y) * in9 ) * K;  // (qc*qf(y))*qf(.)
    o[10] = __builtin_rintf((0.0028228759765625f * xys) * zs) * K; // 2960/2^20
    o[11] = __builtin_rintf((0.45703125f  * y) * in11) * K;
    o[12] = __builtin_rintf((0.373046875f * z) * in12) * K;
    o[13] = __builtin_rintf((0.45703125f  * x) * in11) * K;
    o[14] = __builtin_rintf((1.4453125f   * z) * in14) * K;
    o[15] = __builtin_rintf((0.58984375f  * x) * in15) * K;
}

// Low 32 bits of a generic pointer to __shared__ == wave-relative LDS byte addr.
__device__ __forceinline__ unsigned lds_addr(const void* p) {
    return (unsigned)(unsigned long long)(uintptr_t)p;
}

__global__ __launch_bounds__(BLOCK)
void qsh_kernel(const float* __restrict__ gin, float* __restrict__ gout, int npts) {
    __shared__ __align__(16) float sIn[BLOCK * 3];    //  3 KB
    __shared__ __align__(16) float sOut[BLOCK * 16];  // 16 KB

    const int t = threadIdx.x;
    const long long base = (long long)blockIdx.x * BLOCK;
    const bool full = (base + BLOCK) <= (long long)npts;

    if (full) {
        // ---- stage inputs: 3072 B contiguous, 192 x 16B async chunks (NT) ----
        const int NCHUNK_IN = (BLOCK * 12) / 16;  // 192
        if (t < NCHUNK_IN) {
            unsigned long long g = (unsigned long long)(uintptr_t)
                ((const char*)gin + (size_t)base * 12 + (size_t)t * 16);
            unsigned l = lds_addr((const char*)sIn + t * 16);
            asm volatile("global_load_async_to_lds_b128 %0, %1, off th:TH_LOAD_NT"
                         :: "v"(l), "v"(g) : "memory");
        }
#if __has_builtin(__builtin_amdgcn_s_wait_asynccnt)
        __builtin_amdgcn_s_wait_asynccnt(0);
#else
        asm volatile("s_wait_asynccnt 0" ::: "memory");
#endif
        __syncthreads();

        // ---- compute 16 quantized SH columns for this thread's point ----
        float o[16];
        sh16(sIn[3 * t + 0], sIn[3 * t + 1], sIn[3 * t + 2], o);

        // ---- write AoS tile to LDS (ds_store_b128 x4) ----
        float4* so = (float4*)(sOut + t * 16);
        so[0] = make_float4(o[0],  o[1],  o[2],  o[3]);
        so[1] = make_float4(o[4],  o[5],  o[6],  o[7]);
        so[2] = make_float4(o[8],  o[9],  o[10], o[11]);
        so[3] = make_float4(o[12], o[13], o[14], o[15]);
        __syncthreads();  // barrier waits DScnt -> LDS visible to async engine

        // ---- async store: 16384 B contiguous, 1024 x 16B chunks (NT), 4/thread ----
        #pragma unroll
        for (int k = 0; k < 4; ++k) {
            int c = t + k * BLOCK;
            unsigned long long g = (unsigned long long)(uintptr_t)
                ((char*)gout + (size_t)base * 64 + (size_t)c * 16);
            unsigned l = lds_addr((const char*)sOut + c * 16);
            asm volatile("global_store_async_from_lds_b128 %0, %1, off th:TH_STORE_NT"
                         :: "v"(g), "v"(l) : "memory");
        }
        // ASYNCcnt drained by the implicit wait-idle at s_endpgm.
    } else {
        // tail block (not hit for N = 4,000,000): guarded scalar path
        long long i = base + t;
        if (i < (long long)npts) {
            float o[16];
            sh16(gin[3 * i + 0], gin[3 * i + 1], gin[3 * i + 2], o);
            float4* go = (float4*)(gout + i * 16);
            go[0] = make_float4(o[0],  o[1],  o[2],  o[3]);
            go[1] = make_float4(o[4],  o[5],  o[6],  o[7]);
            go[2] = make_float4(o[8],  o[9],  o[10], o[11]);
            go[3] = make_float4(o[12], o[13], o[14], o[15]);
        }
    }
}

extern "C" void kernel_launch(void* const* d_in, const int* in_sizes, int n_in,
                              void* d_out, int out_size, void* d_ws, size_t ws_size,
                              hipStream_t stream) {
    const float* gin = (const float*)d_in[0];
    float* gout = (float*)d_out;
    const int npts = in_sizes[0] / 3;          // inputs are [N,3] float32
    const int nblk = (npts + BLOCK - 1) / BLOCK;
    qsh_kernel<<<nblk, BLOCK, 0, stream>>>(gin, gout, npts);
}